// OurLayer_40261023433105
// MI455X (gfx1250) — compile-verified
//
#include <hip/hip_runtime.h>

typedef float v2f __attribute__((ext_vector_type(2)));
typedef float v8f __attribute__((ext_vector_type(8)));

#define NB   64
#define CIN  128
#define COUT 128
#define HW   4096   // 64*64
#define HH   64
#define WW   64

// ---------------------------------------------------------------------------
// Kernel 1: per-(n,c) stats. One block per (n*128+c), 256 threads.
// gap[n,c] = mean over HW; pool[n,c,cell] = raw sum over its 16x16 cell.
// float4 coalesced: iter i covers rows [16i,16i+16) so ph == iter, and a
// 4-float chunk never crosses a 16-wide pooling cell.
// ---------------------------------------------------------------------------
__global__ void k_stats(const float* __restrict__ x,
                        float* __restrict__ gap, float* __restrict__ pool) {
    const int nc = blockIdx.x;
    const int t  = threadIdx.x;
    const float4* xp = (const float4*)(x + (size_t)nc * HW);

    __shared__ float cell[16];
    __shared__ float red[256];
    if (t < 16) cell[t] = 0.f;
    __syncthreads();

    const int pw = (t & 15) >> 2;          // ((t*4)%64)/16
    float tot = 0.f;
    float csum[4];
#pragma unroll
    for (int iter = 0; iter < 4; ++iter) {
        float4 v = xp[iter * 256 + t];
        float s = v.x + v.y + v.z + v.w;
        csum[iter] = s;
        tot += s;
    }
#pragma unroll
    for (int iter = 0; iter < 4; ++iter)
        atomicAdd(&cell[iter * 4 + pw], csum[iter]);   // ds_add_f32

    red[t] = tot;
    __syncthreads();
    for (int s = 128; s > 0; s >>= 1) {
        if (t < s) red[t] += red[t + s];
        __syncthreads();
    }
    if (t == 0)  gap[nc] = red[0] * (1.f / 4096.f);
    if (t < 16)  pool[nc * 16 + t] = cell[t];
}

// ---------------------------------------------------------------------------
// WMMA f32 16x16x4 two-layer MLP for one wave (rows [n0,n0+16)):
//   out[16,128] = relu(A[16,128] @ W1[128,8] + b1) @ W2[8,128] + b2
// W1 is pre-staged in LDS zero-padded & transposed: W1t[col(16)][k(128)],
// so B fetches are unconditional contiguous ds_load_b64 (no EXEC juggling
// in the WMMA K-loop). Layouts per CDNA5 ISA 7.12.2:
//   A 16x4:  lane L, vgpr v -> row = L&15, k = v + 2*(L>>4)
//   B 4x16:  lane L, vgpr v -> col = L&15, k = v + 2*(L>>4)
//   C/D:     lane L, vgpr v -> M = v + 8*(L>>4), N = L&15
// ---------------------------------------------------------------------------
__device__ __forceinline__ void mlp_wmma(const float* __restrict__ gap,
                                         const float* __restrict__ W1t,  // LDS [16][128]
                                         const float* __restrict__ b1p,  // LDS [16]
                                         const float* __restrict__ W2,   // global [8][128]
                                         const float* __restrict__ b2,   // global [128]
                                         float* __restrict__ outv,       // global [64][128]
                                         float* __restrict__ hid,        // LDS [16*16]
                                         int n0, int lane) {
    const int row = lane & 15;
    const int col = lane & 15;
    const int kk  = (lane >> 4) << 1;

    // ---- layer 1: [16 x 128] @ [128 x 8(pad16)] ----
    v8f acc = {};
    for (int k0 = 0; k0 < CIN; k0 += 4) {
        v2f a, b;
        a.x = gap[(n0 + row) * CIN + k0 + kk];
        a.y = gap[(n0 + row) * CIN + k0 + kk + 1];
        b.x = W1t[col * CIN + k0 + kk];
        b.y = W1t[col * CIN + k0 + kk + 1];
        acc = __builtin_amdgcn_wmma_f32_16x16x4_f32(
                  false, a, false, b, (short)0, acc, false, false);
    }
    {
        const float bb = b1p[col];                  // already zero-padded
#pragma unroll
        for (int v = 0; v < 8; ++v) {
            const int M = v + ((lane >> 4) << 3);
            hid[M * 16 + col] = fmaxf(acc[v] + bb, 0.f) * ((col < 8) ? 1.f : 0.f);
        }
    }
    __syncthreads();

    // ---- layer 2: [16 x 8] @ [8 x 128], 8 N-tiles ----
    for (int nt = 0; nt < 8; ++nt) {
        v8f acc2 = {};
#pragma unroll
        for (int k0 = 0; k0 < 8; k0 += 4) {
            v2f a, b;
            a.x = hid[row * 16 + k0 + kk];
            a.y = hid[row * 16 + k0 + kk + 1];
            const int nc = nt * 16 + col;
            b.x = W2[(k0 + kk)     * COUT + nc];
            b.y = W2[(k0 + kk + 1) * COUT + nc];
            acc2 = __builtin_amdgcn_wmma_f32_16x16x4_f32(
                       false, a, false, b, (short)0, acc2, false, false);
        }
        const int Ncol = nt * 16 + col;
#pragma unroll
        for (int v = 0; v < 8; ++v) {
            const int M = v + ((lane >> 4) << 3);
            outv[(n0 + M) * COUT + Ncol] = acc2[v] + b2[Ncol];
        }
    }
    __syncthreads();
}

// ---------------------------------------------------------------------------
// Kernel 2: single block, 128 threads (4 waves). xs/sp head + both MLPs.
// ---------------------------------------------------------------------------
__global__ void k_mlp(const float* __restrict__ gap, const float* __restrict__ pool,
                      const float* __restrict__ W_sp, const float* __restrict__ b_sp,
                      const float* __restrict__ W_o1, const float* __restrict__ b_o1,
                      const float* __restrict__ W_o2, const float* __restrict__ b_o2,
                      const float* __restrict__ W_i1, const float* __restrict__ b_i1,
                      const float* __restrict__ W_i2, const float* __restrict__ b_i2,
                      float* __restrict__ sp, float* __restrict__ co,
                      float* __restrict__ ci) {
    __shared__ float xs[NB * 16];
    __shared__ float hid[4][256];
    __shared__ float W1i[16 * CIN];   // zero-padded, transposed [col][k]
    __shared__ float W1o[16 * CIN];
    __shared__ float b1i[16];
    __shared__ float b1o[16];
    const int t = threadIdx.x;

    // stage padded layer-1 weights (removes divergent loads from WMMA loop)
    for (int p = t; p < 16 * CIN; p += 128) {
        const int c = p >> 7, k = p & 127;          // col, k
        W1i[c * CIN + k] = (c < 8) ? W_i1[k * 8 + c] : 0.f;
        W1o[c * CIN + k] = (c < 8) ? W_o1[k * 8 + c] : 0.f;
    }
    if (t < 16) {
        b1i[t] = (t < 8) ? b_i1[t] : 0.f;
        b1o[t] = (t < 8) ? b_o1[t] : 0.f;
    }

    // xs[n,cell] = mean over c of pooled cell means
    for (int p = t; p < NB * 16; p += 128) {
        const int n = p >> 4, cl = p & 15;
        float s = 0.f;
        for (int c = 0; c < CIN; ++c) s += pool[(n * CIN + c) * 16 + cl];
        xs[p] = s * (1.f / (256.f * 128.f));
    }
    __syncthreads();

    // sp[n,9] = xs @ W_sp + b_sp
    for (int p = t; p < NB * 9; p += 128) {
        const int n = p / 9, j = p % 9;
        float s = b_sp[j];
        for (int q = 0; q < 16; ++q) s += xs[n * 16 + q] * W_sp[q * 9 + j];
        sp[p] = s;
    }
    __syncthreads();

    const int wave = t >> 5, lane = t & 31;
    const int n0 = wave << 4;
    mlp_wmma(gap, W1i, b1i, W_i2, b_i2, ci, &hid[wave][0], n0, lane);
    mlp_wmma(gap, W1o, b1o, W_o2, b_o2, co, &hid[wave][0], n0, lane);
}

// ---------------------------------------------------------------------------
// Kernel 3: y[n,hw] = sum_c ci[n,c] * x[n,c,hw].  Bandwidth-bound; x should be
// L2-resident (128 MiB < 192 MB) after k_stats. 512 blocks x 128 thr, float4.
// ---------------------------------------------------------------------------
__global__ void k_ymix(const float* __restrict__ x, const float* __restrict__ ci,
                       float* __restrict__ y) {
    const int n     = blockIdx.x >> 3;
    const int chunk = blockIdx.x & 7;
    const int p4    = chunk * 128 + threadIdx.x;   // float4 index within image
    const float4* xp = (const float4*)(x + (size_t)n * CIN * HW) + p4;
    const float*  cv = ci + n * CIN;

    float4 acc = {0.f, 0.f, 0.f, 0.f};
#pragma unroll 4
    for (int c = 0; c < CIN; ++c) {
        const float s = cv[c];
        const float4 v = xp[(size_t)c * (HW / 4)];
        acc.x += s * v.x; acc.y += s * v.y;
        acc.z += s * v.z; acc.w += s * v.w;
    }
    ((float4*)(y + (size_t)n * HW))[p4] = acc;
}

// ---------------------------------------------------------------------------
// Kernel 4: z = 3x3(sp) over y (zero pad), out[n,co,hw] = co[n,co]*z[hw].
// Block = (n, 4 rows), 256 threads; 128 coalesced row-stores per thread.
// ---------------------------------------------------------------------------
__global__ void k_out(const float* __restrict__ y, const float* __restrict__ sp,
                      const float* __restrict__ co, float* __restrict__ out) {
    const int n  = blockIdx.x >> 4;
    const int rb = blockIdx.x & 15;
    const int t  = threadIdx.x;
    const int h  = rb * 4 + (t >> 6);
    const int w  = t & 63;

    __shared__ float s9[9];
    __shared__ float cov[COUT];
    if (t < 9)    s9[t]  = sp[n * 9 + t];
    if (t < COUT) cov[t] = co[n * COUT + t];
    __syncthreads();

    const float* yp = y + (size_t)n * HW;
    float z = 0.f;
#pragma unroll
    for (int kh = 0; kh < 3; ++kh) {
        const int hh = h + kh - 1;
        if (hh < 0 || hh >= HH) continue;
#pragma unroll
        for (int kw = 0; kw < 3; ++kw) {
            const int ww = w + kw - 1;
            if (ww < 0 || ww >= WW) continue;
            z += s9[kh * 3 + kw] * yp[hh * WW + ww];
        }
    }
    float* op = out + (size_t)n * COUT * HW + h * WW + w;
#pragma unroll 4
    for (int c = 0; c < COUT; ++c) op[(size_t)c * HW] = cov[c] * z;
}

// ---------------------------------------------------------------------------
extern "C" void kernel_launch(void* const* d_in, const int* in_sizes, int n_in,
                              void* d_out, int out_size, void* d_ws, size_t ws_size,
                              hipStream_t stream) {
    const float* x    = (const float*)d_in[0];
    const float* W_sp = (const float*)d_in[1];
    const float* b_sp = (const float*)d_in[2];
    const float* W_o1 = (const float*)d_in[3];
    const float* b_o1 = (const float*)d_in[4];
    const float* W_o2 = (const float*)d_in[5];
    const float* b_o2 = (const float*)d_in[6];
    const float* W_i1 = (const float*)d_in[7];
    const float* b_i1 = (const float*)d_in[8];
    const float* W_i2 = (const float*)d_in[9];
    const float* b_i2 = (const float*)d_in[10];
    float* out = (float*)d_out;

    float* ws   = (float*)d_ws;
    float* gap  = ws;                       // 64*128
    float* pool = gap  + NB * CIN;          // 64*128*16
    float* sp   = pool + NB * CIN * 16;     // 64*9
    float* ci   = sp   + NB * 9;            // 64*128
    float* co   = ci   + NB * CIN;          // 64*128
    float* y    = co   + NB * COUT;         // 64*4096

    k_stats<<<NB * CIN, 256, 0, stream>>>(x, gap, pool);
    k_mlp<<<1, 128, 0, stream>>>(gap, pool, W_sp, b_sp, W_o1, b_o1, W_o2, b_o2,
                                 W_i1, b_i1, W_i2, b_i2, sp, co, ci);
    k_ymix<<<NB * 8, 128, 0, stream>>>(x, ci, y);
    k_out<<<NB * 16, 256, 0, stream>>>(y, sp, co, out);
}